// RelGraphConvHetero_24739011625315
// MI455X (gfx1250) — compile-verified
//
#include <hip/hip_runtime.h>
#include <hip/hip_bf16.h>

// Problem constants (match reference)
#define NN    100000   // nodes
#define EE    160000   // edges per relation
#define RR    4        // relations
#define DD    128      // D_IN == D_OUT
#define LDSW  130      // padded LDS stride for transposed weights (even -> 8B align, conflict-free)

typedef float v2f __attribute__((ext_vector_type(2)));
typedef float v8f __attribute__((ext_vector_type(8)));

// ---------------------------------------------------------------------------
// Stage 2: edge scatter. One wave per edge: lane i atomically adds
// x[src][4i..4i+3] into agg[r][dst][4i..4i+3]; lane 0 bumps deg[r][dst].
// ---------------------------------------------------------------------------
__global__ void __launch_bounds__(256) rgcn_edge_agg(
    const float* __restrict__ x,
    const int*   __restrict__ src,
    const int*   __restrict__ dst,
    float* __restrict__ agg,   // [RR, NN, DD]
    float* __restrict__ deg)   // [RR, NN]
{
    const int lane = threadIdx.x & 31;
    const long wid = ((long)blockIdx.x * blockDim.x + threadIdx.x) >> 5;
    if (wid >= (long)RR * EE) return;

    const int r = (int)(wid / EE);
    const int s = src[wid];
    const int d = dst[wid];

    const float4 v = ((const float4*)(x + (size_t)s * DD))[lane];
    float* ag = agg + ((size_t)r * NN + (size_t)d) * DD + lane * 4;
    atomicAdd(ag + 0, v.x);
    atomicAdd(ag + 1, v.y);
    atomicAdd(ag + 2, v.z);
    atomicAdd(ag + 3, v.w);
    if (lane == 0) atomicAdd(deg + (size_t)r * NN + d, 1.0f);
}

// ---------------------------------------------------------------------------
// Stage 3: fused GEMM  out = relu( x@loopW + sum_r (agg_r/deg_r)@W_r )
// Block = 256 threads = 8 waves; each wave computes a 16-row x 128-col strip
// via v_wmma_f32_16x16x4_f32, accumulating over K = 5*128 (5 segments).
// Segment weights (128x128 f32) staged transposed in LDS per block.
// ---------------------------------------------------------------------------
__global__ void __launch_bounds__(256) rgcn_gemm_wmma(
    const float* __restrict__ x,        // [NN, DD]
    const float* __restrict__ W,        // [RR, DD, DD]
    const float* __restrict__ loopW,    // [DD, DD]
    const float* __restrict__ agg,      // [RR, NN, DD]
    const float* __restrict__ deg,      // [RR, NN]
    float* __restrict__ out)            // [NN, DD]
{
    __shared__ __attribute__((aligned(16))) float lw[DD * LDSW];

    const int tid   = threadIdx.x;
    const int lane  = tid & 31;
    const int wib   = tid >> 5;                 // wave in block (0..7)
    const long strip = (long)blockIdx.x * 8 + wib;
    const long m0    = strip * 16;

    const int half = lane >> 4;                 // 0: K-pair {0,1}; 1: K-pair {2,3}
    const int lrow = lane & 15;                 // row (A) / col (B,D) within tile
    const int koff = half * 2;

    int row  = (int)(m0 + lrow);
    int rowc = row < NN ? row : (NN - 1);       // clamp OOB rows (stores guarded)

    v8f acc[8] = {};                            // 16x128 f32 accumulators

    const float* segsrc[5] = { x,
                               agg,
                               agg + (size_t)1 * NN * DD,
                               agg + (size_t)2 * NN * DD,
                               agg + (size_t)3 * NN * DD };
    const float* segW[5]   = { loopW, W, W + DD * DD, W + 2 * DD * DD, W + 3 * DD * DD };

    for (int seg = 0; seg < 5; ++seg) {
        // --- stage this segment's 128x128 weights transposed into LDS ---
        __syncthreads();
        const float* ws = segW[seg];
        #pragma unroll 4
        for (int i = tid; i < DD * DD; i += 256) {
            const int kk = i >> 7;              // source row (k)
            const int nn = i & 127;             // source col (n)
            lw[nn * LDSW + kk] = ws[i];         // store as [n][k]
        }
        __syncthreads();

        // --- per-row scale: 1/max(deg,1) for relation segments ---
        float inv = 1.0f;
        if (seg > 0)
            inv = 1.0f / fmaxf(deg[(size_t)(seg - 1) * NN + rowc], 1.0f);

        const float* arow = segsrc[seg] + (size_t)rowc * DD;

        for (int k = 0; k < DD; k += 4) {
            v2f a = *(const v2f*)(arow + k + koff);   // A fragment (16x4), per ISA layout
            a.x *= inv;
            a.y *= inv;
            #pragma unroll
            for (int nt = 0; nt < 8; ++nt) {
                // B fragment (4x16): lane%16 selects column, VGPR selects K
                v2f b = *(const v2f*)(&lw[(nt * 16 + lrow) * LDSW + k + koff]);
                acc[nt] = __builtin_amdgcn_wmma_f32_16x16x4_f32(
                    /*neg_a=*/false, a, /*neg_b=*/false, b,
                    /*c_mod=*/(short)0, acc[nt],
                    /*reuse_a=*/false, /*reuse_b=*/false);
            }
        }
    }

    // --- ReLU + store. C/D layout: VGPR i -> row (i + 8*half), lane%16 -> col ---
    #pragma unroll
    for (int nt = 0; nt < 8; ++nt) {
        #pragma unroll
        for (int i = 0; i < 8; ++i) {
            const long rr = m0 + i + half * 8;
            if (rr < NN) {
                float v = acc[nt][i];
                out[(size_t)rr * DD + nt * 16 + lrow] = v > 0.0f ? v : 0.0f;
            }
        }
    }
}

// ---------------------------------------------------------------------------
// Launch: inputs per setup_inputs() order: x, W, loop_weight, src, dst
// ---------------------------------------------------------------------------
extern "C" void kernel_launch(void* const* d_in, const int* in_sizes, int n_in,
                              void* d_out, int out_size, void* d_ws, size_t ws_size,
                              hipStream_t stream) {
    (void)in_sizes; (void)n_in; (void)out_size; (void)ws_size;

    const float* x     = (const float*)d_in[0];
    const float* W     = (const float*)d_in[1];
    const float* loopW = (const float*)d_in[2];
    const int*   src   = (const int*)d_in[3];
    const int*   dst   = (const int*)d_in[4];
    float*       out   = (float*)d_out;

    float* agg = (float*)d_ws;                               // [RR,NN,DD] f32
    float* deg = agg + (size_t)RR * NN * DD;                 // [RR,NN]    f32
    const size_t zero_bytes = ((size_t)RR * NN * DD + (size_t)RR * NN) * sizeof(float);

    // Stage 1: zero accumulators (capturable as a memset node)
    hipMemsetAsync(d_ws, 0, zero_bytes, stream);

    // Stage 2: edge scatter (one wave per edge)
    {
        const long waves  = (long)RR * EE;
        const long blocks = (waves * 32 + 255) / 256;
        rgcn_edge_agg<<<dim3((unsigned)blocks), dim3(256), 0, stream>>>(
            x, src, dst, agg, deg);
    }

    // Stage 3: fused WMMA GEMM + ReLU
    {
        const int strips = (NN + 15) / 16;                   // 6250
        const int blocks = (strips + 7) / 8;                 // 782
        rgcn_gemm_wmma<<<dim3(blocks), dim3(256), 0, stream>>>(
            x, W, loopW, agg, deg, out);
    }
}